// KANCNN_81329500717327
// MI455X (gfx1250) — compile-verified
//
#include <hip/hip_runtime.h>

typedef float v2f __attribute__((ext_vector_type(2)));
typedef float v8f __attribute__((ext_vector_type(8)));

#define OUT_C 64
#define IN_C  64
#define KS    3
#define H     32
#define W     32

// LDS input patch for one 16-pixel tile: [IN_C][3 rows][18 cols]
#define PATCH_COLS 18
#define PATCH_ROWSZ (3 * PATCH_COLS)            // 54 floats per channel
#define PATCH_SIZE (IN_C * PATCH_ROWSZ)         // 3456 floats = 13.8 KB

__global__ __launch_bounds__(256) void kan_conv_wmma(
    const float* __restrict__ x,     // [H][W][IN_C]
    const float* __restrict__ wnum,  // [OUT_C][IN_C][3][3][6]
    const float* __restrict__ wden,  // [OUT_C][IN_C][3][3][4]
    float* __restrict__ out)         // [OUT_C][H][W]
{
    __shared__ float patch[PATCH_SIZE];
    __shared__ float partial[4 * 256];   // c-upper-half wave partials (4 KB)

    const int tid = threadIdx.x;
    const int wg  = blockIdx.x;          // 0..63 : pixel tile (half an image row)
    const int i   = wg >> 1;
    const int j0  = (wg & 1) * 16;

    // ---- stage input patch (zero-padded borders) into LDS ----
    for (int idx = tid; idx < PATCH_SIZE; idx += 256) {
        int c   = idx / PATCH_ROWSZ;
        int rem = idx - c * PATCH_ROWSZ;
        int r   = rem / PATCH_COLS;
        int col = rem - r * PATCH_COLS;
        int ri  = i - 1 + r;
        int cj  = j0 - 1 + col;
        float v = 0.0f;
        if (ri >= 0 && ri < H && cj >= 0 && cj < W)
            v = x[(ri * W + cj) * IN_C + c];
        patch[idx] = v;
    }
    __syncthreads();

    const int wave  = tid >> 5;          // 0..7
    const int lane  = tid & 31;
    const int half  = lane >> 4;         // lane half selects K rows / M rows
    const int p     = lane & 15;         // pixel within tile (N dim) / f row (M dim)
    const int ftile = (wave & 3) * 16;   // f-tile per wave
    const int cbeg  = (wave >> 2) * 32;  // c-split across wave groups
    const int cend  = cbeg + 32;

    v8f acc  = {0.f, 0.f, 0.f, 0.f, 0.f, 0.f, 0.f, 0.f};
    const v8f zero = {0.f, 0.f, 0.f, 0.f, 0.f, 0.f, 0.f, 0.f};

    // Per-lane A-matrix row bases (lane p supplies row f = ftile + p)
    const float* wn_row = wnum + (size_t)(ftile + p) * (IN_C * KS * KS * 6);
    const float* wd_row = wden + (size_t)(ftile + p) * (IN_C * KS * KS * 4);

    for (int c = cbeg; c < cend; ++c) {
        const float* pc   = patch + c * PATCH_ROWSZ;
        const float* wn_c = wn_row + c * (KS * KS * 6);
        const float* wd_c = wd_row + c * (KS * KS * 4);

        // prefetch next channel's weight rows (steady-state L2 stream)
        if (c + 1 < cend) {
            __builtin_prefetch(wn_c + KS * KS * 6, 0, 1);
            __builtin_prefetch(wd_c + KS * KS * 4, 0, 1);
        }

        for (int a = 0; a < KS; ++a) {
            const float* prow = pc + a * PATCH_COLS + p;
            for (int b = 0; b < KS; ++b) {
                const int ab = a * KS + b;

                // per-pixel shifted input value and powers x^1..x^5
                float xv = prow[b];
                float x2 = xv * xv;
                float x3 = x2 * xv;
                float x4 = x2 * x2;
                float x5 = x4 * xv;

                // B matrices (4xN=16): VGPR0 = K0 (lanes 0-15) / K2 (lanes 16-31),
                //                      VGPR1 = K1 / K3
                v2f bp0, bp1, bq;
                bp0.x = half ? x2 : 1.0f;  bp0.y = half ? x3 : xv;   // n0..n3
                bp1.x = half ? 0.f : x4;   bp1.y = half ? 0.f : x5;  // n4,n5,pad,pad
                bq.x  = half ? x3 : xv;    bq.y  = half ? x4 : x2;   // n1..n4

                // A matrices (16x4): lane<16 loads K0,K1; lane>=16 loads K2,K3
                v2f an0 = *(const v2f*)(wn_c + ab * 6 + (half ? 2 : 0)); // n0,n1|n2,n3
                v2f an1 = *(const v2f*)(wn_c + ab * 6 + 4);              // n4,n5
                an1.x = half ? 0.f : an1.x;                              // pad K2,K3
                an1.y = half ? 0.f : an1.y;
                v2f aq  = *(const v2f*)(wd_c + ab * 4 + (half ? 2 : 0)); // b1,b2|b3,b4

                // P = Wnum(16x6) @ pows(6x16) via two chained K=4 WMMAs
                v8f pm = __builtin_amdgcn_wmma_f32_16x16x4_f32(
                    false, an1, false, bp1, (short)0, zero, false, false);
                pm = __builtin_amdgcn_wmma_f32_16x16x4_f32(
                    false, an0, false, bp0, (short)0, pm, false, false);
                // Q = Wden(16x4) @ pows(4x16), single K=4 WMMA
                v8f qm = __builtin_amdgcn_wmma_f32_16x16x4_f32(
                    false, aq, false, bq, (short)0, zero, false, false);

                // rational: acc += P * rcp(1 + |Q|)  (TRANS-pipe reciprocal,
                // co-executes with XDL WMMA; 1 + |Q| >= 1 so rcp is well-conditioned)
#pragma unroll
                for (int v = 0; v < 8; ++v) {
                    float qd = 1.0f + __builtin_fabsf(qm[v]);
                    acc[v] = __builtin_fmaf(pm[v], __builtin_amdgcn_rcpf(qd), acc[v]);
                }
            }
        }
    }

    // ---- deterministic in-block combine of the two c-halves ----
    if (wave >= 4) {
#pragma unroll
        for (int v = 0; v < 8; ++v)
            partial[(wave - 4) * 256 + lane * 8 + v] = acc[v];
    }
    __syncthreads();
    if (wave < 4) {
#pragma unroll
        for (int v = 0; v < 8; ++v)
            acc[v] += partial[wave * 256 + lane * 8 + v];
        // C/D layout: VGPR v holds M = v (lanes 0-15) / M = v+8 (lanes 16-31), N = pixel
#pragma unroll
        for (int v = 0; v < 8; ++v) {
            int f = ftile + v + half * 8;
            out[f * (H * W) + i * W + j0 + p] = acc[v];
        }
    }
}

extern "C" void kernel_launch(void* const* d_in, const int* in_sizes, int n_in,
                              void* d_out, int out_size, void* d_ws, size_t ws_size,
                              hipStream_t stream) {
    const float* x    = (const float*)d_in[0];
    const float* wnum = (const float*)d_in[1];
    const float* wden = (const float*)d_in[2];
    float* out        = (float*)d_out;
    kan_conv_wmma<<<64, 256, 0, stream>>>(x, wnum, wden, out);
}